// SparseConv3DTransposeLayer_11888469476365
// MI455X (gfx1250) — compile-verified
//
#include <hip/hip_runtime.h>

typedef __attribute__((ext_vector_type(2))) float v2f;
typedef __attribute__((ext_vector_type(8))) float v8f;

#define B_   2
#define HI   64
#define WI   64
#define DI   32
#define C_   16
#define F_   8
#define HO   128
#define WO   128
#define DO_  64
#define KELEMS (27 * F_ * C_)   // 3456 floats = 13.8 KB of weights

// One wave computes a 16(do) x 8(F) output tile of one output pixel via
// V_WMMA_F32_16X16X4_F32: A = vals[16 do x 16 C] (masked gather),
// B = W[kh,kw,kd]^T [16 C x 8 F, zero-padded to N=16], K=16 done as 4xK=4.
__global__ __launch_bounds__(512) void sconv3dt_wmma(
    const float* __restrict__ images,
    const int*   __restrict__ base_plane,
    const float* __restrict__ kern,
    float*       __restrict__ out)
{
    __shared__ float sk[KELEMS];
    for (int i = threadIdx.x; i < KELEMS; i += blockDim.x) sk[i] = kern[i];
    __syncthreads();

    const int lane = threadIdx.x & 31;
    const int wave = threadIdx.x >> 5;
    const int tile = blockIdx.x * 16 + wave;       // 16 waves / block

    const int doBase = (tile & 3) * 16;            // 4 do-tiles per pixel
    const int pixel  = tile >> 2;
    const int wo = pixel & (WO - 1);
    const int ho = (pixel >> 7) & (HO - 1);
    const int b  = pixel >> 14;

    const int M    = lane & 15;                    // A row (= do offset) and B col (= F index)
    const int half = lane >> 4;                    // K sub-pair selector per ISA layout

    const int bp_out = base_plane[(b * HO + ho) * WO + wo];
    const int vd = bp_out + doBase + M;            // virtual depth of this lane's do-row

    v8f acc = {};

    #pragma unroll
    for (int kh = 0; kh < 3; ++kh) {
        const int hn = ho + 1 - kh;                            // wave-uniform
        if ((hn & 1) || hn < 0 || hn >= HO) continue;          // uniform skip: EXEC stays full
        const int hi = hn >> 1;
        #pragma unroll
        for (int kw = 0; kw < 3; ++kw) {
            const int wn = wo + 1 - kw;                        // wave-uniform
            if ((wn & 1) || wn < 0 || wn >= WO) continue;
            const int wi = wn >> 1;
            const int bpg = base_plane[(b * HO + hi * 2) * WO + wi * 2] >> 1;
            const float* img = images + (size_t)(((b * HI + hi) * WI + wi) * DI) * C_;
            #pragma unroll
            for (int kd = 0; kd < 3; ++kd) {
                const int dn = vd + 1 - kd;
                const int di = (dn >> 1) - bpg;                // arithmetic shift == floor div
                const bool valid = ((dn & 1) == 0) && (di >= 0) && (di < DI);

                // A fragment: lane holds images[di(M)][c0 + 2*half .. +1] for c0 = 0,4,8,12
                v2f a0 = {}, a1 = {}, a2 = {}, a3 = {};
                if (valid) {
                    const float* row = img + di * C_ + 2 * half;
                    a0 = *(const v2f*)(row + 0);
                    a1 = *(const v2f*)(row + 4);
                    a2 = *(const v2f*)(row + 8);
                    a3 = *(const v2f*)(row + 12);
                }
                // B fragment: W^T[c][f] = kern[kh,kw,kd,f,c]; zero pad f>=8
                v2f b0 = {}, b1 = {}, b2 = {}, b3 = {};
                if (M < F_) {
                    const float* w = sk + ((kh * 3 + kw) * 3 + kd) * (F_ * C_)
                                        + M * C_ + 2 * half;
                    b0 = *(const v2f*)(w + 0);
                    b1 = *(const v2f*)(w + 4);
                    b2 = *(const v2f*)(w + 8);
                    b3 = *(const v2f*)(w + 12);
                }
                // EXEC is all-ones here (divergent loads re-converged)
                acc = __builtin_amdgcn_wmma_f32_16x16x4_f32(false, a0, false, b0, (short)0, acc, false, false);
                acc = __builtin_amdgcn_wmma_f32_16x16x4_f32(false, a1, false, b1, (short)0, acc, false, false);
                acc = __builtin_amdgcn_wmma_f32_16x16x4_f32(false, a2, false, b2, (short)0, acc, false, false);
                acc = __builtin_amdgcn_wmma_f32_16x16x4_f32(false, a3, false, b3, (short)0, acc, false, false);
            }
        }
    }

    // D layout: VGPR r -> M_d = r + 8*half, N = lane&15. Store only N < F_.
    if (M < F_) {
        float* op = out + ((size_t)pixel * DO_ + doBase + 8 * half) * F_ + M;
        #pragma unroll
        for (int r = 0; r < 8; ++r) op[(size_t)r * F_] = acc[r];
    }
}

extern "C" void kernel_launch(void* const* d_in, const int* in_sizes, int n_in,
                              void* d_out, int out_size, void* d_ws, size_t ws_size,
                              hipStream_t stream) {
    const float* images     = (const float*)d_in[0];
    const int*   base_plane = (const int*)d_in[1];
    const float* kern       = (const float*)d_in[2];
    // d_in[3] = default_value: frozen zero scalar per reference; masked taps use 0.
    float* out = (float*)d_out;

    // tiles = B*HO*WO*(DO/16) = 2*128*128*4 = 131072; 16 waves (512 thr) per block
    const int total_tiles = B_ * HO * WO * (DO_ / 16);
    const int blocks = total_tiles / 16;
    hipLaunchKernelGGL(sconv3dt_wmma, dim3(blocks), dim3(512), 0, stream,
                       images, base_plane, kern, out);
}